// LiTLoss_16149077033577
// MI455X (gfx1250) — compile-verified
//
#include <hip/hip_runtime.h>
#include <hip/hip_bf16.h>

typedef __attribute__((ext_vector_type(2))) float v2f;
typedef __attribute__((ext_vector_type(8))) float v8f;

#define SEGS 4          // row-segments for the column pass (parallelism)
#define NMAX 8192

// Cross-half (lane ^ 16) exchange within a wave32 via v_permlanex16_b32:
// pure VALU, no LDS round-trip, no s_wait_dscnt in the hot loop.
__device__ __forceinline__ float xhalf_swap(float v) {
    int r = __builtin_amdgcn_permlanex16(__float_as_int(v), __float_as_int(v),
                                         0x76543210, 0xfedcba98,
                                         /*fi=*/true, /*bound_ctrl=*/false);
    return __int_as_float(r);
}

// ---------------------------------------------------------------------------
// Kernel 1: per-row log-sum-exp (online) + diagonal gather.
// One wave32 per row; lanes stream float4 across the row.
// ---------------------------------------------------------------------------
__global__ void row_lse_kernel(const float* __restrict__ sim,
                               const long long* __restrict__ map,
                               float* __restrict__ rowLSE,
                               float* __restrict__ diag, int N) {
    const int wave = threadIdx.x >> 5;
    const int lane = threadIdx.x & 31;
    const int row  = blockIdx.x * 8 + wave;   // blockDim.x == 256 -> 8 waves
    if (row >= N) return;
    const float* rp = sim + (size_t)row * (size_t)N;

    float m = -INFINITY, s = 0.0f;
    for (int c = lane * 4; c < N; c += 32 * 4) {
        float4 v = *reinterpret_cast<const float4*>(rp + c);
        float x[4] = {v.x, v.y, v.z, v.w};
#pragma unroll
        for (int t = 0; t < 4; ++t) {
            float xv = x[t];
            if (xv > m) {                     // rare after warm-up
                s = s * __expf(m - xv) + 1.0f;
                m = xv;
            } else {
                s += __expf(xv - m);
            }
        }
    }
    // wave32 merge of (max, sum) pairs
#pragma unroll
    for (int off = 16; off > 0; off >>= 1) {
        float m2 = __shfl_xor(m, off, 32);
        float s2 = __shfl_xor(s, off, 32);
        float M  = fmaxf(m, m2);
        s = s * __expf(m - M) + s2 * __expf(m2 - M);
        m = M;
    }
    if (lane == 0) {
        rowLSE[row] = m + __logf(s);
        long long j = map[row];
        diag[row] = rp[j];
    }
}

// ---------------------------------------------------------------------------
// Kernel 2: per-column partial log-sum-exp using V_WMMA_F32_16X16X4_F32.
// Each wave owns 16 columns and one of SEGS row segments. Per iteration it
// covers 16 rows (4 WMMA steps): loads are four contiguous 64B segments per
// 4-row group, exponentials are f32 (v_exp_f32), and
//   C = ones(16x4) x exp_tile(4x16) + C
// accumulates f32 column sums. Each lane's C registers belong to exactly one
// column (lane & 15), so the online-max rescale is a per-lane multiply done
// once per 16 rows. Cross-half chunk max uses v_permlanex16 (no LDS).
// ---------------------------------------------------------------------------
__global__ void col_lse_partial_kernel(const float* __restrict__ sim,
                                       float* __restrict__ pMax,
                                       float* __restrict__ pSum, int N) {
    const int lane = threadIdx.x & 31;
    const int gw   = blockIdx.x * 8 + (threadIdx.x >> 5); // global wave id
    const int seg  = gw & (SEGS - 1);
    const int col0 = (gw / SEGS) * 16;
    if (col0 >= N) return;

    const int n = lane & 15;   // column within the 16-wide group
    const int h = lane >> 4;   // 0: rows r,r+1   1: rows r+2,r+3 (per group)
    const int rowsPerSeg = N / SEGS;
    const int rBeg = seg * rowsPerSeg, rEnd = rBeg + rowsPerSeg;

    const float* base = sim + (size_t)col0 + (size_t)n;
    v2f a; a.x = 1.0f; a.y = 1.0f;            // all-ones A (16x4)
    v8f c = {0.f, 0.f, 0.f, 0.f, 0.f, 0.f, 0.f, 0.f};
    float m = -INFINITY;

    for (int r = rBeg; r < rEnd; r += 16) {
        float x[8];
#pragma unroll
        for (int g = 0; g < 4; ++g) {          // 4-row group g -> one WMMA
            size_t o = (size_t)(r + 4 * g + 2 * h) * (size_t)N;
            x[2 * g]     = base[o];
            x[2 * g + 1] = base[o + (size_t)N];
        }
        // chunk max over 16 rows for this lane's column
        float cm = x[0];
#pragma unroll
        for (int t = 1; t < 8; ++t) cm = fmaxf(cm, x[t]);
        cm = fmaxf(cm, xhalf_swap(cm));        // merge lane pair l, l^16

        float newm  = fmaxf(m, cm);
        float scale = __expf(m - newm);        // ==1 if max unchanged; 0 on first iter
#pragma unroll
        for (int t = 0; t < 8; ++t) c[t] *= scale;
        m = newm;

#pragma unroll
        for (int g = 0; g < 4; ++g) {
            v2f b;
            b.x = __expf(x[2 * g]     - m);
            b.y = __expf(x[2 * g + 1] - m);
            // D[mrow][n] = C[mrow][n] + sum_k B[k][n]  -> f32 column sums
            c = __builtin_amdgcn_wmma_f32_16x16x4_f32(false, a, false, b,
                                                      (short)0, c, false, false);
        }
    }

    if (lane < 16) {
        pMax[(size_t)seg * N + col0 + lane] = m;
        pSum[(size_t)seg * N + col0 + lane] = c[0];
    }
}

// ---------------------------------------------------------------------------
// Kernel 3: merge column partials (in LDS), then final scalar reduction.
// ---------------------------------------------------------------------------
__global__ void finalize_kernel(const float* __restrict__ rowLSE,
                                const float* __restrict__ diag,
                                const float* __restrict__ pMax,
                                const float* __restrict__ pSum,
                                const long long* __restrict__ map,
                                float* __restrict__ out, int N) {
    __shared__ float colLSE[NMAX];
    const int tid = threadIdx.x;

    for (int c1 = tid; c1 < N; c1 += 256) {
        float M = -INFINITY;
#pragma unroll
        for (int s1 = 0; s1 < SEGS; ++s1)
            M = fmaxf(M, pMax[(size_t)s1 * N + c1]);
        float S = 0.0f;
#pragma unroll
        for (int s1 = 0; s1 < SEGS; ++s1)
            S += pSum[(size_t)s1 * N + c1] * __expf(pMax[(size_t)s1 * N + c1] - M);
        colLSE[c1] = M + __logf(S);
    }
    __syncthreads();

    float a1 = 0.0f, a2 = 0.0f;
    for (int i = tid; i < N; i += 256) {
        float d = diag[i];
        a1 += rowLSE[i] - d;
        a2 += colLSE[(int)map[i]] - d;
    }
    float v = a1 + a2;
#pragma unroll
    for (int off = 16; off > 0; off >>= 1) v += __shfl_xor(v, off, 32);

    __shared__ float wsum[8];
    if ((tid & 31) == 0) wsum[tid >> 5] = v;
    __syncthreads();
    if (tid < 8) {
        float t = wsum[tid];
#pragma unroll
        for (int off = 4; off > 0; off >>= 1) t += __shfl_xor(t, off, 8);
        if (tid == 0) out[0] = 0.5f * t / (float)N;
    }
}

// ---------------------------------------------------------------------------
extern "C" void kernel_launch(void* const* d_in, const int* in_sizes, int n_in,
                              void* d_out, int out_size, void* d_ws, size_t ws_size,
                              hipStream_t stream) {
    const float*     sim = (const float*)d_in[0];
    const long long* map = (const long long*)d_in[1]; // int64 mapping
    float* out = (float*)d_out;
    const int N = in_sizes[1];                        // 8192

    float* ws     = (float*)d_ws;
    float* rowLSE = ws;                  // N
    float* diag   = ws + N;              // N
    float* pMax   = ws + 2 * N;          // SEGS*N
    float* pSum   = ws + (2 + SEGS) * N; // SEGS*N   (total (2+2*SEGS)*N floats = 320KB)

    // Kernel 1: one wave per row, 8 waves per block.
    row_lse_kernel<<<N / 8, 256, 0, stream>>>(sim, map, rowLSE, diag, N);

    // Kernel 2: (N/16)*SEGS waves, 8 waves per block.
    int waves2 = (N / 16) * SEGS;
    col_lse_partial_kernel<<<waves2 / 8, 256, 0, stream>>>(sim, pMax, pSum, N);

    // Kernel 3: single block merges partials and produces the scalar.
    finalize_kernel<<<1, 256, 0, stream>>>(rowLSE, diag, pMax, pSum, map, out, N);
}